// DeepJMTModel_83047487635954
// MI455X (gfx1250) — compile-verified
//
#include <hip/hip_runtime.h>
#include <hip/hip_bf16.h>
#include <math.h>

// ---------------- problem constants ----------------
#define HDIM   256
#define INDIM  8
#define PPOI   50000
#define NNODE  8192
#define NHEAD  4
#define NFEAT  4
#define HIDD   4
#define NCLS   2
#define ALPHA  0.2f

// ---------------- output layout (float elements) ----------------
#define OUT_H1    0
#define OUT_PH    256
#define OUT_QHH2  512
#define OUT_AH2   66048
#define OUT_IDX   131584
#define OUT_RAW   131591

// ---------------- workspace layout (float elements) ----------------
// h1, cL, cP contiguous so mL = ws[0..767]
#define WS_H1      0
#define WS_CL      256
#define WS_CP      512
#define WS_NPH     768
#define WS_DEXP    1024
#define WS_S1T     51200    /* s1h[k*8192+n]            (head-major) */
#define WS_T1T     83968    /* t1h[k*8192+n]            (head-major) */
#define WS_HH      116736   /* hhh[(k*8192+n)*4+c]      (head-major) */
#define WS_HEADOUT 247808   /* headout[n*16+k*4+c]      (node-major) */
#define WS_HO      378880
#define WS_S2      395264
#define WS_T2      403456
#define WS_DOTG    411648
#define WS_MLMM    419840
#define WS_ANW     419968
#define WS_MASK    428032   /* uint32 bitset: 8192*256 words = 8MB */

typedef float v2f __attribute__((ext_vector_type(2)));
typedef float v8f __attribute__((ext_vector_type(8)));
typedef _Float16 v16h __attribute__((ext_vector_type(16)));

__device__ __forceinline__ float wredsum(float v) {
#pragma unroll
  for (int o = 16; o > 0; o >>= 1) v += __shfl_down(v, o, 32);
  return v;
}
__device__ __forceinline__ float sigmoidf_(float x) { return 1.0f / (1.0f + __expf(-x)); }
__device__ __forceinline__ float leakyf_(float x) { return x > 0.0f ? x : ALPHA * x; }
__device__ __forceinline__ float eluf_(float x) { return x > 0.0f ? x : expm1f(x); }

// Async-stage an 8-row mask tile (8*256 u32 = 8KB) into LDS via the CDNA5
// async-to-LDS path (ASYNCcnt-tracked), then s_wait_asynccnt + barrier.
__device__ __forceinline__ void stage_mask8(const unsigned* __restrict__ maskw,
                                            int r0, unsigned* smask) {
#if defined(__AMDGCN__)
  const unsigned* gbase = maskw + (size_t)r0 * 256;
#pragma unroll
  for (int t = 0; t < 2; ++t) {
    int q = threadIdx.x + t * 256;               // 0..511 chunks of 16B
    unsigned lds_off = (unsigned)(size_t)&smask[q * 4];
    const unsigned* ga = gbase + q * 4;
    asm volatile("global_load_async_to_lds_b128 %0, %1, off"
                 :: "v"(lds_off), "v"(ga) : "memory");
  }
  asm volatile("s_wait_asynccnt 0x0" ::: "memory");
#else
  for (int t = 0; t < 8; ++t)
    smask[t * 256 + threadIdx.x] = maskw[(size_t)(r0 + t) * 256 + threadIdx.x];
#endif
  __syncthreads();
}

// ---------------- GRU cell: hout = GRU(concat(xa,xb), h) ----------------
__global__ void k_gru(const float* __restrict__ xa, int la,
                      const float* __restrict__ xb, int lb,
                      const float* __restrict__ h,
                      const float* __restrict__ Wih, const float* __restrict__ Whh,
                      const float* __restrict__ bih, const float* __restrict__ bhh,
                      float* __restrict__ hout, float* __restrict__ hout2) {
  __shared__ float xin[16];
  __shared__ float hsh[HDIM];
  int tid = threadIdx.x;
  int L = la + lb;
  if (tid < la) xin[tid] = xa[tid];
  else if (tid < L) xin[tid] = xb[tid - la];
  hsh[tid] = h[tid];
  __syncthreads();

  int i = tid;
  float gir = bih[i], giz = bih[HDIM + i], gin = bih[2 * HDIM + i];
  for (int f = 0; f < L; ++f) {
    float xf = xin[f];
    gir += xf * Wih[i * L + f];
    giz += xf * Wih[(HDIM + i) * L + f];
    gin += xf * Wih[(2 * HDIM + i) * L + f];
  }
  float ghr = bhh[i], ghz = bhh[HDIM + i], ghn = bhh[2 * HDIM + i];
  for (int j = 0; j < HDIM; ++j) {
    float hj = hsh[j];
    ghr += hj * Whh[i * HDIM + j];
    ghz += hj * Whh[(HDIM + i) * HDIM + j];
    ghn += hj * Whh[(2 * HDIM + i) * HDIM + j];
  }
  float r = sigmoidf_(gir + ghr);
  float z = sigmoidf_(giz + ghz);
  float n = tanhf(gin + r * ghn);
  float ho = (1.0f - z) * n + z * hsh[i];
  hout[i] = ho;
  if (hout2) hout2[i] = ho;
}

__global__ void k_copy(const float* __restrict__ s, float* __restrict__ d, int n) {
  int i = blockIdx.x * blockDim.x + threadIdx.x;
  if (i < n) d[i] = s[i];
}

// ---------------- d = exp(-dist/2) ----------------
__global__ void k_dexp(const float* __restrict__ dist, float* __restrict__ dexp) {
  int i = blockIdx.x * blockDim.x + threadIdx.x;
  if (i < PPOI) dexp[i] = __expf(-0.5f * dist[i]);
}

// ---------------- qhi / qhh2 / aH2 via WMMA outer product ----------------
// One wave per 16x16 tile of the 256x256 outer(h1, nph); exp/div applied to
// the accumulator registers. 256 tiles -> 32 blocks x 8 waves.
__global__ void k_qmat(const float* __restrict__ h1, const float* __restrict__ nph,
                       const float* __restrict__ qhh, const float* __restrict__ aH,
                       float* __restrict__ out_qhh2, float* __restrict__ out_aH2) {
  int wave = (blockIdx.x * blockDim.x + threadIdx.x) >> 5;  // 0..255
  int lane = threadIdx.x & 31;
  int i0 = (wave >> 4) * 16, j0 = (wave & 15) * 16;
  int mb = (lane < 16) ? 0 : 8;
  int col = j0 + (lane & 15);
  float dv[8];
#if __has_builtin(__builtin_amdgcn_wmma_f32_16x16x4_f32)
  float hv = h1[i0 + (lane & 15)];
  float nv = nph[j0 + (lane & 15)];
  v2f a; a[0] = (lane < 16) ? hv : 0.0f; a[1] = 0.0f;
  v2f b; b[0] = (lane < 16) ? nv : 0.0f; b[1] = 0.0f;
  v8f c = {};
  v8f d = __builtin_amdgcn_wmma_f32_16x16x4_f32(false, a, false, b, (short)0, c, false, false);
#pragma unroll
  for (int v = 0; v < 8; ++v) dv[v] = d[v];
#else
#pragma unroll
  for (int v = 0; v < 8; ++v) dv[v] = h1[i0 + v + mb] * nph[col];
#endif
#pragma unroll
  for (int v = 0; v < 8; ++v) {
    int idx = (i0 + v + mb) * HDIM + col;
    float qhi = __expf(dv[v]);
    float q2 = qhh[idx] + qhi;
    out_qhh2[idx] = q2;
    out_aH2[idx] = aH[idx] + qhi / q2;
  }
}

// ---------------- cP = aH2 @ nph ----------------
__global__ void k_cP(const float* __restrict__ aH2, const float* __restrict__ nph,
                     float* __restrict__ cP) {
  __shared__ float nsh[HDIM];
  int i = threadIdx.x;
  nsh[i] = nph[i];
  __syncthreads();
  float s = 0.0f;
  for (int j = 0; j < HDIM; ++j) s += aH2[i * HDIM + j] * nsh[j];
  cP[i] = s;
}

// ---------------- cL: weighted POI barycenter per hidden unit ----------------
__global__ void k_cL(const float* __restrict__ h1, const float* __restrict__ w,
                     const float* __restrict__ loc, const float* __restrict__ dexp,
                     float* __restrict__ cL) {
  int i = blockIdx.x;  // 0..255
  float a = w[0] * h1[i];
  float se0 = 0.f, sl0 = 0.f, se1 = 0.f, sl1 = 0.f;
  for (int p = threadIdx.x; p < PPOI; p += 256) {
    float dp = dexp[p];
    float l0 = loc[2 * p], l1 = loc[2 * p + 1];
    float e0 = __expf(a * l0 * dp);
    float e1 = __expf(a * l1 * dp);
    se0 += e0; sl0 += e0 * l0;
    se1 += e1; sl1 += e1 * l1;
  }
  __shared__ float red[8 * 4];
  int wid = threadIdx.x >> 5, lane = threadIdx.x & 31;
  float v0 = wredsum(se0), v1 = wredsum(sl0), v2 = wredsum(se1), v3 = wredsum(sl1);
  if (lane == 0) { red[wid*4] = v0; red[wid*4+1] = v1; red[wid*4+2] = v2; red[wid*4+3] = v3; }
  __syncthreads();
  if (threadIdx.x == 0) {
    float t0=0,t1=0,t2=0,t3=0;
    for (int wv = 0; wv < 8; ++wv) { t0+=red[wv*4]; t1+=red[wv*4+1]; t2+=red[wv*4+2]; t3+=red[wv*4+3]; }
    cL[i] = t1 / t0 + t3 / t2;
  }
}

// ---------------- adj (256MB int32) -> 8MB bitset, read exactly once ----------------
__global__ void k_mask(const int* __restrict__ adj, unsigned* __restrict__ maskw) {
  long long tid = (long long)blockIdx.x * blockDim.x + threadIdx.x;  // 64M threads
  int pred = adj[tid] > 0;
  unsigned long long b = __ballot(pred);  // wave32: low 32 bits valid
  if ((threadIdx.x & 31) == 0) maskw[tid >> 5] = (unsigned)b;
}

// ---------------- GAT precompute, head-major outputs ----------------
__global__ void k_gat_pre(const float* __restrict__ nf, const float* __restrict__ W1,
                          const float* __restrict__ a1, float* __restrict__ hhh,
                          float* __restrict__ s1h, float* __restrict__ t1h) {
  int n = blockIdx.x * blockDim.x + threadIdx.x;
  if (n >= NNODE) return;
  float f0 = nf[n*4], f1 = nf[n*4+1], f2 = nf[n*4+2], f3 = nf[n*4+3];
#pragma unroll
  for (int k = 0; k < NHEAD; ++k) {
    const float* W = W1 + k * 16;  // [f][c]
    const float* a = a1 + k * 8;
    float hv[4];
#pragma unroll
    for (int c = 0; c < 4; ++c)
      hv[c] = f0 * W[c] + f1 * W[4 + c] + f2 * W[8 + c] + f3 * W[12 + c];
    float s = hv[0]*a[0] + hv[1]*a[1] + hv[2]*a[2] + hv[3]*a[3];
    float t = hv[0]*a[4] + hv[1]*a[5] + hv[2]*a[6] + hv[3]*a[7];
    size_t base = ((size_t)k * NNODE + n);
#pragma unroll
    for (int c = 0; c < 4; ++c) hhh[base * 4 + c] = hv[c];
    s1h[base] = s;
    t1h[base] = t;
  }
}

// ---------------- GAT layer 1: 8 rows x 1 head per block ----------------
// L2-traffic-optimal: 20B/j streamed per block (vs 80B/j unfused), mask tile
// staged via async-to-LDS. 1024 row-tiles x 4 heads = 4096 blocks.
__global__ void k_att1(const float* __restrict__ s1h, const float* __restrict__ t1h,
                       const float* __restrict__ hhh, const unsigned* __restrict__ maskw,
                       float* __restrict__ headout) {
  __shared__ unsigned smask[8 * 256];
  __shared__ float red[8 * 40];
  int k = blockIdx.x & 3;
  int r0 = (blockIdx.x >> 2) * 8;
  stage_mask8(maskw, r0, smask);

  const float* sp = s1h + (size_t)k * NNODE;
  const float* tp = t1h + (size_t)k * NNODE;
  const float4* hp = (const float4*)hhh + (size_t)k * NNODE;
  float si[8];
#pragma unroll
  for (int r = 0; r < 8; ++r) si[r] = sp[r0 + r];

  float den[8];
  float num[32];
#pragma unroll
  for (int r = 0; r < 8; ++r) den[r] = 0.f;
#pragma unroll
  for (int q = 0; q < 32; ++q) num[q] = 0.f;

  for (int j = threadIdx.x; j < NNODE; j += 256) {
    float tj = tp[j];
    float4 h4 = hp[j];
    int wq = j >> 5, bp = j & 31;
#pragma unroll
    for (int r = 0; r < 8; ++r) {
      if ((smask[r * 256 + wq] >> bp) & 1u) {
        float p = __expf(leakyf_(si[r] + tj));
        den[r] += p;
        num[r*4+0] += p * h4.x;
        num[r*4+1] += p * h4.y;
        num[r*4+2] += p * h4.z;
        num[r*4+3] += p * h4.w;
      }
    }
  }
  int wid = threadIdx.x >> 5, lane = threadIdx.x & 31;
#pragma unroll
  for (int r = 0; r < 8; ++r) {
    float v = wredsum(den[r]);
    if (lane == 0) red[wid*40 + r] = v;
  }
#pragma unroll
  for (int q = 0; q < 32; ++q) {
    float v = wredsum(num[q]);
    if (lane == 0) red[wid*40 + 8 + q] = v;
  }
  __syncthreads();
  if (threadIdx.x == 0) {
#pragma unroll
    for (int r = 0; r < 8; ++r) {
      float d = 0.f;
      for (int wv = 0; wv < 8; ++wv) d += red[wv*40 + r];
#pragma unroll
      for (int c = 0; c < 4; ++c) {
        float s = 0.f;
        for (int wv = 0; wv < 8; ++wv) s += red[wv*40 + 8 + r*4 + c];
        float v = (d > 0.f) ? s / d : 0.f;
        headout[(size_t)(r0 + r) * 16 + k * 4 + c] = eluf_(v);
      }
    }
  }
}

// ---------------- GAT output projection ----------------
__global__ void k_gat_out_pre(const float* __restrict__ headout, const float* __restrict__ Wout,
                              const float* __restrict__ aout, float* __restrict__ ho,
                              float* __restrict__ s2, float* __restrict__ t2) {
  int n = blockIdx.x * blockDim.x + threadIdx.x;
  if (n >= NNODE) return;
  const float* hc = headout + (size_t)n * 16;
  float o0 = 0.f, o1 = 0.f;
#pragma unroll
  for (int q = 0; q < 16; ++q) {
    float h = hc[q];
    o0 += h * Wout[q*2];
    o1 += h * Wout[q*2 + 1];
  }
  ho[n*2] = o0; ho[n*2 + 1] = o1;
  s2[n] = o0 * aout[0] + o1 * aout[1];
  t2[n] = o0 * aout[2] + o1 * aout[3];
}

// ---------------- GAT layer 2: 8 rows per block + log_softmax + dotg ----------------
__global__ void k_att2(const float* __restrict__ s2, const float* __restrict__ t2,
                       const float* __restrict__ ho, const unsigned* __restrict__ maskw,
                       const float* __restrict__ nf, float* __restrict__ dotg) {
  __shared__ unsigned smask[8 * 256];
  __shared__ float red[8 * 24];
  int r0 = blockIdx.x * 8;
  stage_mask8(maskw, r0, smask);

  float si[8];
#pragma unroll
  for (int r = 0; r < 8; ++r) si[r] = s2[r0 + r];
  float den[8], n0[8], n1[8];
#pragma unroll
  for (int r = 0; r < 8; ++r) { den[r] = 0.f; n0[r] = 0.f; n1[r] = 0.f; }

  const float2* hop = (const float2*)ho;
  for (int j = threadIdx.x; j < NNODE; j += 256) {
    float tj = t2[j];
    float2 h2 = hop[j];
    int wq = j >> 5, bp = j & 31;
#pragma unroll
    for (int r = 0; r < 8; ++r) {
      if ((smask[r * 256 + wq] >> bp) & 1u) {
        float p = __expf(leakyf_(si[r] + tj));
        den[r] += p; n0[r] += p * h2.x; n1[r] += p * h2.y;
      }
    }
  }
  int wid = threadIdx.x >> 5, lane = threadIdx.x & 31;
#pragma unroll
  for (int r = 0; r < 8; ++r) {
    float a = wredsum(den[r]);
    float b = wredsum(n0[r]);
    float c = wredsum(n1[r]);
    if (lane == 0) { red[wid*24 + r] = a; red[wid*24 + 8 + r] = b; red[wid*24 + 16 + r] = c; }
  }
  __syncthreads();
  if (threadIdx.x == 0) {
#pragma unroll
    for (int r = 0; r < 8; ++r) {
      float td = 0.f, t0 = 0.f, t1 = 0.f;
      for (int wv = 0; wv < 8; ++wv) {
        td += red[wv*24 + r]; t0 += red[wv*24 + 8 + r]; t1 += red[wv*24 + 16 + r];
      }
      float v0 = eluf_((td > 0.f) ? t0 / td : 0.f);
      float v1 = eluf_((td > 0.f) ? t1 / td : 0.f);
      float m = fmaxf(v0, v1);
      float lse = m + logf(__expf(v0 - m) + __expf(v1 - m));
      float g0 = v0 - lse, g1 = v1 - lse;
      int i = r0 + r;
      dotg[i] = g0 * nf[i*4] + g1 * nf[i*4 + 1];
    }
  }
}

// ---------------- max/min of mL (768) ----------------
__global__ void k_mlmm(const float* __restrict__ mL, float* __restrict__ mm) {
  float mx = -INFINITY, mn = INFINITY;
  for (int f = threadIdx.x; f < 768; f += 256) {
    float v = mL[f];
    mx = fmaxf(mx, v); mn = fminf(mn, v);
  }
#pragma unroll
  for (int o = 16; o > 0; o >>= 1) {
    mx = fmaxf(mx, __shfl_down(mx, o, 32));
    mn = fminf(mn, __shfl_down(mn, o, 32));
  }
  __shared__ float rx[8], rn[8];
  int wid = threadIdx.x >> 5, lane = threadIdx.x & 31;
  if (lane == 0) { rx[wid] = mx; rn[wid] = mn; }
  __syncthreads();
  if (threadIdx.x == 0) {
    float MX = rx[0], MN = rn[0];
    for (int wv = 1; wv < 8; ++wv) { MX = fmaxf(MX, rx[wv]); MN = fminf(MN, rn[wv]); }
    mm[0] = MX; mm[1] = MN;
  }
}

// ---------------- raw = outer(dotg, mL) via V_WMMA_F32_16X16X4_F32 ----------------
__global__ void k_raw(const float* __restrict__ dotg, const float* __restrict__ mL,
                      float* __restrict__ raw) {
  int wave = (blockIdx.x * blockDim.x + threadIdx.x) >> 5;  // 0..24575
  int lane = threadIdx.x & 31;
  int tr = wave / 48, tc = wave % 48;
  int n0 = tr * 16, f0 = tc * 16;
  int mb = (lane < 16) ? 0 : 8;
  int col = f0 + (lane & 15);
#if __has_builtin(__builtin_amdgcn_wmma_f32_16x16x4_f32)
  float dv = dotg[n0 + (lane & 15)];   // unconditional load; select below (no branch)
  float mv = mL[col];
  v2f a; a[0] = (lane < 16) ? dv : 0.0f; a[1] = 0.0f;
  v2f b; b[0] = (lane < 16) ? mv : 0.0f; b[1] = 0.0f;
  v8f c = {};
  v8f d = __builtin_amdgcn_wmma_f32_16x16x4_f32(false, a, false, b, (short)0, c, false, false);
#pragma unroll
  for (int v = 0; v < 8; ++v) raw[(size_t)(n0 + v + mb) * 768 + col] = d[v];
#else
  float mlv = mL[col];
  for (int v = 0; v < 8; ++v) {
    int row = n0 + v + mb;
    raw[(size_t)row * 768 + col] = dotg[row] * mlv;
  }
#endif
}

// ---------------- anw[n] = max_f dotg[n]*mL[f] (analytic) ----------------
__global__ void k_anw(const float* __restrict__ dotg, const float* __restrict__ mm,
                      float* __restrict__ anw) {
  int n = blockIdx.x * blockDim.x + threadIdx.x;
  if (n >= NNODE) return;
  float d = dotg[n];
  anw[n] = (d >= 0.f) ? d * mm[0] : d * mm[1];
}

// ---------------- top-7 argmax (softmax monotone -> same indices) ----------------
__global__ void k_topk(const float* __restrict__ anw, float* __restrict__ out_idx) {
  __shared__ float sv[NNODE];
  __shared__ float rv[256];
  __shared__ int   ri[256];
  for (int n = threadIdx.x; n < NNODE; n += 256) sv[n] = anw[n];
  __syncthreads();
  for (int t = 0; t < 7; ++t) {
    float bv = -INFINITY; int bi = 0x7fffffff;
    for (int n = threadIdx.x; n < NNODE; n += 256) {
      float v = sv[n];
      if (v > bv || (v == bv && n < bi)) { bv = v; bi = n; }
    }
    rv[threadIdx.x] = bv; ri[threadIdx.x] = bi;
    __syncthreads();
    for (int s = 128; s > 0; s >>= 1) {
      if (threadIdx.x < s) {
        float ov = rv[threadIdx.x + s]; int oi = ri[threadIdx.x + s];
        if (ov > rv[threadIdx.x] || (ov == rv[threadIdx.x] && oi < ri[threadIdx.x])) {
          rv[threadIdx.x] = ov; ri[threadIdx.x] = oi;
        }
      }
      __syncthreads();
    }
    if (threadIdx.x == 0) {
      out_idx[t] = (float)ri[0];
      sv[ri[0]] = -INFINITY;
    }
    __syncthreads();
  }
}

// ================================================================
extern "C" void kernel_launch(void* const* d_in, const int* in_sizes, int n_in,
                              void* d_out, int out_size, void* d_ws, size_t ws_size,
                              hipStream_t stream) {
  const float* x        = (const float*)d_in[0];
  const float* nextHid  = (const float*)d_in[1];
  const float* user     = (const float*)d_in[2];
  const float* periodHid= (const float*)d_in[3];
  const float* qhh      = (const float*)d_in[4];
  const float* aH       = (const float*)d_in[5];
  const float* poi_loc  = (const float*)d_in[6];
  const float* poi_dist = (const float*)d_in[7];
  const float* node_feat= (const float*)d_in[8];
  const int*   adj      = (const int*)d_in[9];
  (void)d_in[10]; // pre == 0
  const float* weight   = (const float*)d_in[11];
  const float* g1_Wih   = (const float*)d_in[12];
  const float* g1_Whh   = (const float*)d_in[13];
  const float* g1_bih   = (const float*)d_in[14];
  const float* g1_bhh   = (const float*)d_in[15];
  const float* g3_Wih   = (const float*)d_in[16];
  const float* g3_Whh   = (const float*)d_in[17];
  const float* g3_bih   = (const float*)d_in[18];
  const float* g3_bhh   = (const float*)d_in[19];
  const float* gat_W1   = (const float*)d_in[20];
  const float* gat_a1   = (const float*)d_in[21];
  const float* gat_Wout = (const float*)d_in[22];
  const float* gat_aout = (const float*)d_in[23];

  float* out = (float*)d_out;
  float* wsf = (float*)d_ws;
  unsigned* wsm = (unsigned*)(wsf + WS_MASK);

  // GRUs (tiny) + periodHid passthrough
  k_gru<<<1, 256, 0, stream>>>(x, INDIM, nullptr, 0, nextHid,
                               g1_Wih, g1_Whh, g1_bih, g1_bhh,
                               wsf + WS_H1, out + OUT_H1);
  k_gru<<<1, 256, 0, stream>>>(user, 1, x, INDIM, periodHid,
                               g3_Wih, g3_Whh, g3_bih, g3_bhh,
                               wsf + WS_NPH, nullptr);
  k_copy<<<1, 256, 0, stream>>>(periodHid, out + OUT_PH, HDIM);

  // qhh2 / aH2 (WMMA outer product) / cP
  k_qmat<<<32, 256, 0, stream>>>(wsf + WS_H1, wsf + WS_NPH, qhh, aH,
                                 out + OUT_QHH2, out + OUT_AH2);
  k_cP<<<1, 256, 0, stream>>>(out + OUT_AH2, wsf + WS_NPH, wsf + WS_CP);

  // POI barycenter cL
  k_dexp<<<(PPOI + 255) / 256, 256, 0, stream>>>(poi_dist, wsf + WS_DEXP);
  k_cL<<<256, 256, 0, stream>>>(wsf + WS_H1, weight, poi_loc, wsf + WS_DEXP, wsf + WS_CL);

  // adj -> bitset (single 256MB HBM read; GAT then runs out of 8MB bitset in L2)
  k_mask<<<(NNODE * (NNODE / 256)), 256, 0, stream>>>(adj, wsm);

  // GAT
  k_gat_pre<<<NNODE / 256, 256, 0, stream>>>(node_feat, gat_W1, gat_a1,
                                             wsf + WS_HH, wsf + WS_S1T, wsf + WS_T1T);
  k_att1<<<(NNODE / 8) * NHEAD, 256, 0, stream>>>(wsf + WS_S1T, wsf + WS_T1T,
                                                  wsf + WS_HH, wsm, wsf + WS_HEADOUT);
  k_gat_out_pre<<<NNODE / 256, 256, 0, stream>>>(wsf + WS_HEADOUT, gat_Wout, gat_aout,
                                                 wsf + WS_HO, wsf + WS_S2, wsf + WS_T2);
  k_att2<<<NNODE / 8, 256, 0, stream>>>(wsf + WS_S2, wsf + WS_T2, wsf + WS_HO,
                                        wsm, node_feat, wsf + WS_DOTG);

  // raw = outer(dotg, mL) via WMMA; anw analytic; top-7
  k_mlmm<<<1, 256, 0, stream>>>(wsf /* mL = [h1,cL,cP] */, wsf + WS_MLMM);
  k_raw<<<3072, 256, 0, stream>>>(wsf + WS_DOTG, wsf, out + OUT_RAW);
  k_anw<<<NNODE / 256, 256, 0, stream>>>(wsf + WS_DOTG, wsf + WS_MLMM, wsf + WS_ANW);
  k_topk<<<1, 256, 0, stream>>>(wsf + WS_ANW, out + OUT_IDX);
}